// Decoder_3461743640648
// MI455X (gfx1250) — compile-verified
//
#include <hip/hip_runtime.h>

typedef __attribute__((ext_vector_type(2))) float v2f;
typedef __attribute__((ext_vector_type(8))) float v8f;

#define BN_EPS 1e-5f
#define STATS_ROWS 512

// ---------------------------------------------------------------------------
// Pass 1: per-block partial sum / sum-of-squares per channel.
// Layout of partial: [block][0:C]=sum, [block][C:2C]=sumsq
// Deterministic: each thread owns fixed channels, LDS reduce in fixed order.
// ---------------------------------------------------------------------------
__global__ __launch_bounds__(256) void stats_partial_k(
    const float* __restrict__ f, int N, int C, float* __restrict__ partial)
{
    __shared__ float sm[512];
    const int tid      = threadIdx.x;
    const int cThreads = (C < 256) ? C : 256;   // threads covering channels
    const int rgroups  = 256 / cThreads;        // 1, 2 or 4 row groups
    const int cpt      = (C > 256) ? 2 : 1;     // channels per thread (C=512)
    const int laneC    = tid % cThreads;
    const int rg       = tid / cThreads;
    const int r0       = blockIdx.x * STATS_ROWS;
    const int r1       = (r0 + STATS_ROWS < N) ? (r0 + STATS_ROWS) : N;

    float s0 = 0.f, q0 = 0.f, s1 = 0.f, q1 = 0.f;
    for (int r = r0 + rg; r < r1; r += rgroups) {
        float v = f[(size_t)r * C + laneC];
        s0 += v; q0 += v * v;
        if (cpt == 2) {
            float w = f[(size_t)r * C + laneC + 256];
            s1 += w; q1 += w * w;
        }
    }
    sm[tid] = s0; sm[256 + tid] = q0;
    __syncthreads();
    if (rg == 0) {
        for (int g = 1; g < rgroups; ++g) {           // fixed order -> deterministic
            s0 += sm[tid + g * cThreads];
            q0 += sm[256 + tid + g * cThreads];
        }
        float* p = partial + (size_t)blockIdx.x * 2 * C;
        p[laneC]     = s0;
        p[C + laneC] = q0;
        if (cpt == 2) { p[laneC + 256] = s1; p[C + laneC + 256] = q1; }
    }
}

// ---------------------------------------------------------------------------
// Pass 2: reduce partials (sequential, deterministic), produce per-channel
// BN affine: ab[c] = a_c = gamma*rsqrt(var+eps), ab[C+c] = b_c = beta - mean*a_c
// ---------------------------------------------------------------------------
__global__ __launch_bounds__(256) void stats_finalize_k(
    const float* __restrict__ partial, int nblk, int N, int C,
    const float* __restrict__ gamma, const float* __restrict__ beta,
    float* __restrict__ ab)
{
    int c = blockIdx.x * 256 + threadIdx.x;
    if (c >= C) return;
    float S = 0.f, Q = 0.f;
    for (int b = 0; b < nblk; ++b) {
        S += partial[(size_t)b * 2 * C + c];
        Q += partial[(size_t)b * 2 * C + C + c];
    }
    float inv  = 1.f / (float)N;
    float mean = S * inv;
    float var  = Q * inv - mean * mean;   // biased variance
    float a    = gamma[c] * rsqrtf(var + BN_EPS);
    ab[c]      = a;
    ab[C + c]  = beta[c] - mean * a;
}

// ---------------------------------------------------------------------------
// Fold BN scale into the weight, K-pair-interleaved layout so a B fragment is
// one aligned 8-byte load:  Wtp[(k/2)*512 + 2*j + (k&1)] = W[j,k] * a_k
// ---------------------------------------------------------------------------
__global__ __launch_bounds__(256) void fold_weight_k(
    const float* __restrict__ W, const float* __restrict__ ab,
    int C, float* __restrict__ Wtp)
{
    int c = blockIdx.x;       // 0..C-1
    int j = threadIdx.x;      // 0..255
    Wtp[(size_t)(c >> 1) * 512 + 2 * j + (c & 1)] = W[(size_t)j * C + c] * ab[c];
}

// bias[j] = sum_c b_c * W[j,c]
__global__ __launch_bounds__(256) void fold_bias_k(
    const float* __restrict__ W, const float* __restrict__ ab,
    int C, float* __restrict__ bias)
{
    int j = threadIdx.x;
    float s = 0.f;
    for (int c = 0; c < C; ++c) s += ab[C + c] * W[(size_t)j * C + c];
    bias[j] = s;
}

// ---------------------------------------------------------------------------
// GEMM: out[n,j] = sum_c f[n,c]*W'[j,c] + bias[j]  (+ fused gather-residuals
// on the finest level). One block = 256 threads = 8 waves; block covers
// ROWS x 256 outputs. Wave w owns cols [32w, 32w+32) and all ROWS/16 row
// tiles. A tile staged in LDS (stride C+4 floats -> conflict-free frag reads).
// V_WMMA_F32_16X16X4_F32 fragments:
//   A: lane L -> row L%16, K pair at (L/16)*2      (2 f32 regs)
//   B: lane L -> col L%16, K pair at (L/16)*2      (2 f32 regs, one b64 load)
//   D: reg v  -> row v + 8*(L/16), col L%16        (8 f32 regs)
// ---------------------------------------------------------------------------
template <int C, int ROWS>
__global__ __launch_bounds__(256) void gemm_bn_wmma(
    const float* __restrict__ f, int N,
    const float* __restrict__ Wtp, const float* __restrict__ bias,
    float* __restrict__ out,
    const float* __restrict__ g0, const int* __restrict__ idx0,
    const float* __restrict__ g1, const int* __restrict__ idx1,
    const float* __restrict__ g2, const int* __restrict__ idx2)
{
    constexpr int RT     = ROWS / 16;
    constexpr int STRIDE = C + 4;            // bank-conflict padding
    __shared__ float As[ROWS * STRIDE];

    const int tid  = threadIdx.x;
    const int row0 = blockIdx.x * ROWS;

    // Stage A tile (coalesced float4 loads, zero-fill tail rows).
    for (int e = tid; e < ROWS * C / 4; e += 256) {
        int r   = (e * 4) / C;
        int col = (e * 4) % C;
        float4 v = make_float4(0.f, 0.f, 0.f, 0.f);
        if (row0 + r < N)
            v = *reinterpret_cast<const float4*>(f + (size_t)(row0 + r) * C + col);
        *reinterpret_cast<float4*>(&As[r * STRIDE + col]) = v;
    }
    __syncthreads();

    const int wave  = tid >> 5;
    const int lane  = tid & 31;
    const int lhalf = lane >> 4;
    const int lmod  = lane & 15;
    const int col0  = wave * 32;

    v8f acc[RT][2];
#pragma unroll
    for (int rt = 0; rt < RT; ++rt) {
        acc[rt][0] = (v8f){0.f, 0.f, 0.f, 0.f, 0.f, 0.f, 0.f, 0.f};
        acc[rt][1] = (v8f){0.f, 0.f, 0.f, 0.f, 0.f, 0.f, 0.f, 0.f};
    }

    // Per-lane base into the pair-interleaved B panel: pair row (kk/2 + lhalf),
    // column (col0 + lmod).  Each fragment = one aligned 8-byte load.
    const float* wbase = Wtp + (size_t)lhalf * 512 + 2 * (col0 + lmod);

    for (int kk = 0; kk < C; kk += 4) {
        const int kb = kk + lhalf * 2;
        const float* wb = wbase + (size_t)(kk >> 1) * 512;
        v2f b0 = *reinterpret_cast<const v2f*>(wb);        // col tile 0: K=kb,kb+1
        v2f b1 = *reinterpret_cast<const v2f*>(wb + 32);   // col tile 1
#pragma unroll
        for (int rt = 0; rt < RT; ++rt) {
            v2f a = *reinterpret_cast<const v2f*>(&As[(rt * 16 + lmod) * STRIDE + kb]);
            acc[rt][0] = __builtin_amdgcn_wmma_f32_16x16x4_f32(
                false, a, false, b0, (short)0, acc[rt][0], false, false);
            acc[rt][1] = __builtin_amdgcn_wmma_f32_16x16x4_f32(
                false, a, false, b1, (short)0, acc[rt][1], false, false);
        }
    }

    const bool finalMode = (idx0 != nullptr);
#pragma unroll
    for (int rt = 0; rt < RT; ++rt) {
#pragma unroll
        for (int ct = 0; ct < 2; ++ct) {
            const int j  = col0 + ct * 16 + lmod;
            const float bj = bias[j];
#pragma unroll
            for (int v = 0; v < 8; ++v) {
                const int r = row0 + rt * 16 + lhalf * 8 + v;
                if (r < N) {
                    float val = acc[rt][ct][v] + bj;
                    if (finalMode) {
                        val += g0[(size_t)idx0[r] * 256 + j];
                        val += g1[(size_t)idx1[r] * 256 + j];
                        val += g2[(size_t)idx2[r] * 256 + j];
                    }
                    out[(size_t)r * 256 + j] = val;
                }
            }
        }
    }
}

// ---------------------------------------------------------------------------
extern "C" void kernel_launch(void* const* d_in, const int* in_sizes, int n_in,
                              void* d_out, int out_size, void* d_ws, size_t ws_size,
                              hipStream_t stream)
{
    (void)in_sizes; (void)n_in; (void)out_size; (void)ws_size;

    const int Ns[4] = {3750, 15000, 60000, 240000};
    const int Cs[4] = {512, 256, 128, 64};

    const float* f[4]; const float* gamma[4]; const float* beta[4]; const float* W[4];
    for (int i = 0; i < 4; ++i) {
        f[i]     = (const float*)d_in[4 * i + 0];
        gamma[i] = (const float*)d_in[4 * i + 1];
        beta[i]  = (const float*)d_in[4 * i + 2];
        W[i]     = (const float*)d_in[4 * i + 3];
    }
    const int* idx[3];
    for (int i = 0; i < 3; ++i) idx[i] = (const int*)d_in[16 + i];

    // Workspace layout (floats)
    float* ws = (float*)d_ws;
    size_t o = 0;
    float* g0 = ws + o; o += (size_t)3750  * 256;
    float* g1 = ws + o; o += (size_t)15000 * 256;
    float* g2 = ws + o; o += (size_t)60000 * 256;
    float* ab[4]; float* Wtp[4]; float* bias[4]; float* part[4]; int nblk[4];
    for (int i = 0; i < 4; ++i) {
        nblk[i] = (Ns[i] + STATS_ROWS - 1) / STATS_ROWS;
        ab[i]   = ws + o; o += (size_t)2 * Cs[i];
        Wtp[i]  = ws + o; o += (size_t)Cs[i] * 256;
        bias[i] = ws + o; o += 256;
        part[i] = ws + o; o += (size_t)nblk[i] * 2 * Cs[i];
    }

    // BN statistics + weight folding for all levels.
    for (int i = 0; i < 4; ++i) {
        stats_partial_k<<<nblk[i], 256, 0, stream>>>(f[i], Ns[i], Cs[i], part[i]);
        stats_finalize_k<<<(Cs[i] + 255) / 256, 256, 0, stream>>>(
            part[i], nblk[i], Ns[i], Cs[i], gamma[i], beta[i], ab[i]);
        fold_weight_k<<<Cs[i], 256, 0, stream>>>(W[i], ab[i], Cs[i], Wtp[i]);
        fold_bias_k<<<1, 256, 0, stream>>>(W[i], ab[i], Cs[i], bias[i]);
    }

    // Coarse-level projections into workspace.
    gemm_bn_wmma<512, 16><<<(3750 + 15) / 16, 256, 0, stream>>>(
        f[0], 3750, Wtp[0], bias[0], g0,
        nullptr, nullptr, nullptr, nullptr, nullptr, nullptr);
    gemm_bn_wmma<256, 32><<<(15000 + 31) / 32, 256, 0, stream>>>(
        f[1], 15000, Wtp[1], bias[1], g1,
        nullptr, nullptr, nullptr, nullptr, nullptr, nullptr);
    gemm_bn_wmma<128, 64><<<(60000 + 63) / 64, 256, 0, stream>>>(
        f[2], 60000, Wtp[2], bias[2], g2,
        nullptr, nullptr, nullptr, nullptr, nullptr, nullptr);

    // Finest level: GEMM with fused gather-residual epilogue -> d_out.
    gemm_bn_wmma<64, 64><<<(240000 + 63) / 64, 256, 0, stream>>>(
        f[3], 240000, Wtp[3], bias[3], (float*)d_out,
        g0, idx[0], g1, idx[1], g2, idx[2]);
}